// DynamicMaskHead_38869454029679
// MI455X (gfx1250) — compile-verified
//
#include <hip/hip_runtime.h>
#include <math.h>

#define IN_CH 10
#define HW 25600                 // 160*160
#define NPARAMS 169
#define BLOCKS_PER_INST 8
#define THREADS 256
#define WAVES 8
#define PIX_PER_BLOCK (HW / BLOCKS_PER_INST)          // 3200
#define TILES_PER_WAVE (PIX_PER_BLOCK / (WAVES * 16)) // 25

typedef __attribute__((ext_vector_type(2))) float v2f;
typedef __attribute__((ext_vector_type(8))) float v8f;

// D = A(16x4 f32) * B(4x16 f32) + C(16x16 f32)
__device__ __forceinline__ v8f wmma4(v2f a, v2f b, v8f c) {
  return __builtin_amdgcn_wmma_f32_16x16x4_f32(false, a, false, b, (short)0, c,
                                               false, false);
}

// single-instruction ReLU: v_med3_f32(x, 0, FLT_MAX)
__device__ __forceinline__ float relu(float x) {
  return __builtin_amdgcn_fmed3f(x, 0.0f, 3.402823466e38f);
}

// swap lane halves (lane <-> lane^16) via ds_swizzle SWAPX16
__device__ __forceinline__ float swap16(float x) {
  return __int_as_float(__builtin_amdgcn_ds_swizzle(__float_as_int(x), 0x401f));
}

__global__ __launch_bounds__(THREADS) void dyn_mask_head_kernel(
    const float* __restrict__ params, const float* __restrict__ features,
    float* __restrict__ out) {
  // A tiles with channel replication into padded rows:
  //  w1s rows: 0=ch0 1=ch1 4=ch4 5=ch5 | 8=ch2 9=ch3 12=ch6 13=ch7 | rest 0
  //            (cols 10,11 always 0)
  //  w2s rows: 0..3 = out-ch 0..3 | 8..11 = out-ch 4..7 | rest 0
  __shared__ float w1s[16 * 12];
  __shared__ float w2s[16 * 8];
  __shared__ float w3s[8];
  __shared__ float b1s[8];
  __shared__ float b2s[8];
  __shared__ float b3s;

  const int tid = threadIdx.x;
  const int n = blockIdx.x / BLOCKS_PER_INST;
  const int chunk = blockIdx.x % BLOCKS_PER_INST;

  for (int i = tid; i < 16 * 12; i += THREADS) w1s[i] = 0.f;
  for (int i = tid; i < 16 * 8; i += THREADS) w2s[i] = 0.f;
  __syncthreads();

  const float* p = params + (size_t)n * NPARAMS;
  if (tid < 17) {
    // One standardized filter row per thread: 8 rows of w1, 8 of w2, 1 of w3.
    int cin;
    const float* src;
    float* dst;
    if (tid < 8) {
      // layer-1 channel tid -> replicated row placement
      int r1 = tid + ((tid & 2) ? 6 : 0);  // 0,1,8,9,4,5,12,13
      cin = 10; src = p + tid * 10; dst = w1s + r1 * 12;
    } else if (tid < 16) {
      int oc = tid - 8;
      int r2 = (oc < 4) ? oc : oc + 4;     // 0..3, 8..11
      cin = 8; src = p + 80 + oc * 8; dst = w2s + r2 * 8;
    } else {
      cin = 8; src = p + 144; dst = w3s;
    }
    float m = 0.f;
    for (int c = 0; c < cin; ++c) m += src[c];
    m /= (float)cin;
    float v = 0.f;
    for (int c = 0; c < cin; ++c) { float t = src[c] - m; v += t * t; }
    float s = sqrtf(v / (float)(cin - 1)) + 1e-5f;  // unbiased std (ddof=1)
    for (int c = 0; c < cin; ++c) dst[c] = (src[c] - m) / s;
  } else if (tid < 25) {
    b1s[tid - 17] = p[152 + (tid - 17)];
  } else if (tid < 33) {
    b2s[tid - 25] = p[160 + (tid - 25)];
  } else if (tid == 33) {
    b3s = p[168];
  }
  __syncthreads();

  const int lane = tid & 31;
  const int wave = tid >> 5;
  const int half = lane >> 4;  // 0: lanes 0-15, 1: lanes 16-31
  const int row = lane & 15;

  // Loop-invariant A fragments.
  // A layout (16x4 f32): row M = lane&15 for both halves; lanes 0-15 hold
  // K = k0,k0+1; lanes 16-31 hold K = k0+2,k0+3.
  v2f a1[3], a2[2];
#pragma unroll
  for (int c = 0; c < 3; ++c) {
    int k = c * 4 + 2 * half;
    a1[c].x = w1s[row * 12 + k];
    a1[c].y = w1s[row * 12 + k + 1];
  }
#pragma unroll
  for (int c = 0; c < 2; ++c) {
    int k = c * 4 + 2 * half;
    a2[c].x = w2s[row * 8 + k];
    a2[c].y = w2s[row * 8 + k + 1];
  }

  // Bias accumulators matching the replicated-row layout.
  // D1 v_j: low half = row j, high half = row j+8.
  v8f c1 = {};
  c1[0] = half ? b1s[2] : b1s[0];
  c1[1] = half ? b1s[3] : b1s[1];
  c1[4] = half ? b1s[6] : b1s[4];
  c1[5] = half ? b1s[7] : b1s[5];
  v8f c2 = {};
#pragma unroll
  for (int j = 0; j < 4; ++j) c2[j] = half ? b2s[j + 4] : b2s[j];

  // Layer-3: low half reduces ch0-3, high half reduces ch4-7.
  float w3L[4];
#pragma unroll
  for (int j = 0; j < 4; ++j) w3L[j] = half ? w3s[j + 4] : w3s[j];
  const float accInit = half ? 0.f : b3s;  // b3 contributed once (low half)

  const float* f = features + (size_t)n * IN_CH * HW;
  float* o = out + (size_t)n * HW;

  // Two 32-bit element indices so every load/store can use
  // vgpr-index + sgpr-base (scale_offset) + immediate channel offset.
  int pix = chunk * PIX_PER_BLOCK + wave * 16 + row;  // ch8/ch9 + store index
  int idxA = (2 * half) * HW + pix;                   // half-dependent channels

  for (int t = 0; t < TILES_PER_WAVE; ++t) {
    // Layer-1 B fragments (4x16 f32): v.x = row k0+2*half, v.y = row k0+2*half+1.
    v2f b0, b1f, b2f;
    b0.x = f[idxA];
    b0.y = f[idxA + 1 * HW];
    b1f.x = f[idxA + 4 * HW];
    b1f.y = f[idxA + 5 * HW];
    // chunk k0=8: upper half pairs with zero A columns 10,11, so re-loading
    // ch8/9 there is harmless (and stays in bounds).
    b2f.x = f[pix + 8 * HW];
    b2f.y = f[pix + 9 * HW];

    v8f d = wmma4(a1[0], b0, c1);
    d = wmma4(a1[1], b1f, d);
    d = wmma4(a1[2], b2f, d);

    // Thanks to channel replication, D1 already IS the layer-2 B layout:
    // e0: low=(ch0,ch1) high=(ch2,ch3); e1: low=(ch4,ch5) high=(ch6,ch7).
    v2f e0, e1;
    e0.x = relu(d[0]);
    e0.y = relu(d[1]);
    e1.x = relu(d[4]);
    e1.y = relu(d[5]);

    v8f d2 = wmma4(a2[0], e0, c2);
    d2 = wmma4(a2[1], e1, d2);

    // Layer 3: split dot product across lane halves, combine with one swap.
    float acc = accInit;
#pragma unroll
    for (int j = 0; j < 4; ++j) acc = fmaf(w3L[j], relu(d2[j]), acc);
    float tot = acc + swap16(acc);
    // Both lane halves hold the identical total for pixel `pix` (row=lane&15),
    // so an unconditional store is safe: duplicate lanes write the same dword.
    o[pix] = tot;

    pix += WAVES * 16;
    idxA += WAVES * 16;
  }
}

extern "C" void kernel_launch(void* const* d_in, const int* in_sizes, int n_in,
                              void* d_out, int out_size, void* d_ws,
                              size_t ws_size, hipStream_t stream) {
  const float* params = (const float*)d_in[0];
  const float* features = (const float*)d_in[1];
  float* out = (float*)d_out;
  const int n_inst = in_sizes[0] / NPARAMS;  // 200
  dim3 grid(n_inst * BLOCKS_PER_INST);
  dim3 block(THREADS);
  dyn_mask_head_kernel<<<grid, block, 0, stream>>>(params, features, out);
}